// local_self_attention_20366734918016
// MI455X (gfx1250) — compile-verified
//
#include <hip/hip_runtime.h>
#include <math.h>

#define BB 8
#define CC 32
#define NN 4096
#define KK 32
#define HH 4
#define DD 8
#define LN_EPS 1e-5f
#define ATT_EPS 1e-6f

typedef __attribute__((ext_vector_type(2))) float v2f;
typedef __attribute__((ext_vector_type(8))) float v8f;

__device__ __forceinline__ float elu1f(float x) { return x > 0.f ? x + 1.f : __expf(x); }

// C(16x16) += A(16xk) * B(kx16), operands streamed from LDS with arbitrary strides.
// A[m][k] at pa[m*a_rs + k*a_cs]; B[k][n] at pb[k*b_rs + n*b_cs].
// f32 WMMA layout (ISA 7.12.2): A/B: lane&15 = M/N, K = k0 + 2*(lane>>4) + vgpr.
__device__ __forceinline__ v8f wmma_f32_16x16(const float* pa, int a_rs, int a_cs,
                                              const float* pb, int b_rs, int b_cs,
                                              int kdim, v8f acc) {
  const int lane = threadIdx.x & 31;
  const int mn = lane & 15;
  const int kh = (lane >> 4) << 1;
  for (int k0 = 0; k0 < kdim; k0 += 4) {
    v2f a, b;
    a.x = pa[mn * a_rs + (k0 + kh + 0) * a_cs];
    a.y = pa[mn * a_rs + (k0 + kh + 1) * a_cs];
    b.x = pb[(k0 + kh + 0) * b_rs + mn * b_cs];
    b.y = pb[(k0 + kh + 1) * b_rs + mn * b_cs];
    acc = __builtin_amdgcn_wmma_f32_16x16x4_f32(false, a, false, b, (short)0, acc,
                                                false, false);
  }
  return acc;
}

// ---------------------------------------------------------------------------
// Kernel A: transpose feat -> xt[B,N,C], norms xx, Qf = elu1((xt+posmlp(xyz))@w_q)
// ---------------------------------------------------------------------------
__global__ __launch_bounds__(256) void prep_kernel(
    const float* __restrict__ feat, const float* __restrict__ xyz,
    const float* __restrict__ w_pos1, const float* __restrict__ b_pos1,
    const float* __restrict__ w_pos2, const float* __restrict__ b_pos2,
    const float* __restrict__ w_q,
    float* __restrict__ xt, float* __restrict__ xx, float* __restrict__ qf) {
  __shared__ float s_wp1[3 * CC], s_bp1[CC], s_wp2[CC * CC], s_bp2[CC], s_wq[CC * CC];
  for (int t = threadIdx.x; t < 3 * CC; t += 256) s_wp1[t] = w_pos1[t];
  for (int t = threadIdx.x; t < CC; t += 256) { s_bp1[t] = b_pos1[t]; s_bp2[t] = b_pos2[t]; }
  for (int t = threadIdx.x; t < CC * CC; t += 256) { s_wp2[t] = w_pos2[t]; s_wq[t] = w_q[t]; }
  __syncthreads();

  const int p = blockIdx.x * 256 + threadIdx.x;
  const int b = p / NN, n = p % NN;

  float row[CC];
  float s = 0.f;
  #pragma unroll
  for (int c = 0; c < CC; ++c) {
    float v = feat[((size_t)b * CC + c) * NN + n];
    row[c] = v;
    s += v * v;
    xt[(size_t)p * CC + c] = v;
  }
  xx[p] = s;

  const float X = xyz[(size_t)p * 3 + 0];
  const float Y = xyz[(size_t)p * 3 + 1];
  const float Z = xyz[(size_t)p * 3 + 2];
  float z1[CC];
  #pragma unroll
  for (int j = 0; j < CC; ++j)
    z1[j] = fmaxf(0.f, s_bp1[j] + X * s_wp1[j] + Y * s_wp1[CC + j] + Z * s_wp1[2 * CC + j]);
  float qin[CC];
  for (int c = 0; c < CC; ++c) {
    float acc = s_bp2[c];
    #pragma unroll
    for (int j = 0; j < CC; ++j) acc += z1[j] * s_wp2[j * CC + c];
    qin[c] = row[c] + acc;
  }
  for (int c = 0; c < CC; ++c) {
    float acc = 0.f;
    #pragma unroll
    for (int k = 0; k < CC; ++k) acc += qin[k] * s_wq[k * CC + c];
    qf[(size_t)p * CC + c] = elu1f(acc);
  }
}

// ---------------------------------------------------------------------------
// Kernel B: feature-space KNN with WMMA Gram tiles.
// Block = 128 thr (4 waves) = 64 query rows of one batch; wave owns 16 rows.
// Candidates staged 64 rows at a time in LDS (shared by all waves), scores
// 2*Xq.Xc^T computed via v_wmma_f32_16x16x4_f32 (B read transposed from the
// row-major candidate tile), norms folded in while scattering C-tiles to a
// padded score buffer; lanes 0..15 keep per-row insertion-sorted top-32.
// ---------------------------------------------------------------------------
#define QROWS 64   // query rows per block
#define CBLK 64    // candidate rows staged per iteration
__global__ __launch_bounds__(128) void knn_kernel(const float* __restrict__ xt,
                                                  const float* __restrict__ xx,
                                                  int* __restrict__ knn) {
  __shared__ float s_q[QROWS * 33];
  __shared__ float s_qxx[QROWS];
  __shared__ float s_c[CBLK * 33];
  __shared__ float s_cxx[CBLK];
  __shared__ float s_sc[4][16 * 66];   // per-wave 16x64 padded score tile

  const int tid = threadIdx.x, lane = tid & 31, w = tid >> 5;
  const int b = blockIdx.x >> 6;              // NN/QROWS == 64 groups per batch
  const int q0 = (blockIdx.x & 63) * QROWS;
  const float* xb = xt + (size_t)b * NN * CC;
  const float* xxb = xx + (size_t)b * NN;

  for (int t = tid; t < QROWS * CC; t += 128) {
    const int r = t >> 5, c = t & 31;
    s_q[r * 33 + c] = xb[(size_t)(q0 + r) * CC + c];
  }
  for (int t = tid; t < QROWS; t += 128) s_qxx[t] = xxb[q0 + t];

  float best[KK];
  int bidx[KK];
  #pragma unroll
  for (int j = 0; j < KK; ++j) { best[j] = -3.0e38f; bidx[j] = 0; }

  for (int m0 = 0; m0 < NN; m0 += CBLK) {
    __syncthreads();
    for (int t = tid; t < CBLK * CC; t += 128) {
      const int r = t >> 5, c = t & 31;
      s_c[r * 33 + c] = xb[(size_t)(m0 + r) * CC + c];
    }
    for (int t = tid; t < CBLK; t += 128) s_cxx[t] = xxb[m0 + t];
    __syncthreads();

    // prefetch next candidate block into caches (global_prefetch_b8)
    if (m0 + CBLK < NN)
      __builtin_prefetch((const char*)(xb + (size_t)(m0 + CBLK) * CC) + tid * 64, 0, 3);

    // scores for this wave's 16 query rows vs 64 candidates
    for (int nt = 0; nt < CBLK; nt += 16) {
      v8f a = {};
      a = wmma_f32_16x16(&s_q[(w * 16) * 33], 33, 1, &s_c[nt * 33], 1, 33, CC, a);
      const int cn = lane & 15, mr0 = (lane >> 4) << 3;
      #pragma unroll
      for (int v = 0; v < 8; ++v) {
        const int mr = mr0 + v;
        s_sc[w][mr * 66 + nt + cn] =
            2.f * a[v] - s_qxx[w * 16 + mr] - s_cxx[nt + cn];
      }
    }
    __syncthreads();

    if (lane < 16) { // per-row top-k update over the 64 fresh scores
      const int r = lane;
      for (int j = 0; j < CBLK; ++j) {
        const float pd = s_sc[w][r * 66 + j];
        if (pd > best[KK - 1]) {
          int t = KK - 1;
          while (t > 0 && best[t - 1] < pd) {
            best[t] = best[t - 1];
            bidx[t] = bidx[t - 1];
            --t;
          }
          best[t] = pd;
          bidx[t] = m0 + j;
        }
      }
    }
  }

  if (lane < 16) {
    const int q = q0 + w * 16 + lane;
    int* kp = knn + ((size_t)b * NN + q) * KK;
    #pragma unroll
    for (int j = 0; j < KK; ++j) kp[j] = bidx[j];
  }
}

// ---------------------------------------------------------------------------
// Kernel C: per-point linear attention. 1 wave == 1 point, 2 waves per WG.
// kv_in built in LDS (lane == neighbor), K/V projections + Kf^T@Vn via WMMA.
// ---------------------------------------------------------------------------
__global__ __launch_bounds__(64) void attn_kernel(
    const float* __restrict__ xt, const float* __restrict__ xyz,
    const int* __restrict__ knn, const float* __restrict__ qf,
    const float* __restrict__ w_pos1, const float* __restrict__ b_pos1,
    const float* __restrict__ w_pos2, const float* __restrict__ b_pos2,
    const float* __restrict__ w_k, const float* __restrict__ w_v,
    float* __restrict__ msg) {
  __shared__ float s_wp1[96], s_bp1[32], s_wp2[1024], s_bp2[32];
  __shared__ float s_wk[32 * 33], s_wv[32 * 33];
  __shared__ float s_kv[2][32 * 33], s_kf[2][32 * 33], s_vn[2][32 * 33], s_ks[2][32];

  const int tid = threadIdx.x, lane = tid & 31, w = tid >> 5;
  for (int t = tid; t < 96; t += 64) s_wp1[t] = w_pos1[t];
  for (int t = tid; t < 32; t += 64) { s_bp1[t] = b_pos1[t]; s_bp2[t] = b_pos2[t]; }
  for (int t = tid; t < 1024; t += 64) {
    const int r = t >> 5, c = t & 31;
    s_wp2[t] = w_pos2[t];
    s_wk[r * 33 + c] = w_k[t];
    s_wv[r * 33 + c] = w_v[t];
  }
  const int p = blockIdx.x * 2 + w;
  const int b = p / NN;
  __syncthreads();

  { // gather neighbor (lane == j), posmlp, build kv_in row in LDS
    const int gi = knn[(size_t)p * KK + lane];
    const float* fr = xt + ((size_t)b * NN + gi) * CC;
    const float* xr = xyz + ((size_t)b * NN + gi) * 3;
    const float X = xr[0], Y = xr[1], Z = xr[2];
    float z1[32];
    #pragma unroll
    for (int j = 0; j < 32; ++j)
      z1[j] = fmaxf(0.f, s_bp1[j] + X * s_wp1[j] + Y * s_wp1[32 + j] + Z * s_wp1[64 + j]);
    for (int c = 0; c < 32; ++c) {
      float acc = s_bp2[c];
      #pragma unroll
      for (int j = 0; j < 32; ++j) acc += z1[j] * s_wp2[j * 32 + c];
      s_kv[w][lane * 33 + c] = fr[c] + acc;
    }
  }
  __syncthreads();

  // Kf = elu1(kv @ w_k), Vn = (kv @ w_v) / K    (WMMA, 4 tiles each)
  for (int mt = 0; mt < 32; mt += 16)
    for (int nt = 0; nt < 32; nt += 16) {
      v8f aK = {};
      aK = wmma_f32_16x16(&s_kv[w][mt * 33], 33, 1, &s_wk[nt], 33, 1, 32, aK);
      v8f aV = {};
      aV = wmma_f32_16x16(&s_kv[w][mt * 33], 33, 1, &s_wv[nt], 33, 1, 32, aV);
      const int cn = lane & 15, m0 = (lane >> 4) << 3;
      #pragma unroll
      for (int v = 0; v < 8; ++v) {
        s_kf[w][(mt + m0 + v) * 33 + nt + cn] = elu1f(aK[v]);
        s_vn[w][(mt + m0 + v) * 33 + nt + cn] = aV[v] * (1.f / (float)KK);
      }
    }
  __syncthreads();

  { // Ksum per channel (lane == channel)
    float ks = 0.f;
    #pragma unroll
    for (int j = 0; j < 32; ++j) ks += s_kf[w][j * 33 + lane];
    s_ks[w][lane] = ks;
  }
  // KVfull = Kf^T @ Vn (diagonal 8x8 head blocks are what we need) -> reuse s_kv
  for (int mt = 0; mt < 32; mt += 16)
    for (int nt = 0; nt < 32; nt += 16) {
      v8f a = {};
      a = wmma_f32_16x16(&s_kf[w][mt], 1, 33, &s_vn[w][nt], 33, 1, 32, a);
      const int cn = lane & 15, m0 = (lane >> 4) << 3;
      #pragma unroll
      for (int v = 0; v < 8; ++v) s_kv[w][(mt + m0 + v) * 33 + nt + cn] = a[v];
    }
  __syncthreads();

  { // msg[c] = (Qf[h,:] . KV[h][:,v]) / (Qf[h,:] . Ksum[h,:] + eps) * K
    const float* qrow = qf + (size_t)p * CC;
    const int h = lane >> 3;
    float num = 0.f, den = ATT_EPS;
    #pragma unroll
    for (int d = 0; d < DD; ++d) {
      const float qv = qrow[h * DD + d];
      num += qv * s_kv[w][(h * DD + d) * 33 + lane];
      den += qv * s_ks[w][h * DD + d];
    }
    msg[(size_t)p * CC + lane] = num / den * (float)KK;
  }
}

// ---------------------------------------------------------------------------
// Kernel D: merge proj + LN1 + concat + MLP + LN2 + residual + transpose out.
// 1 wave == 16 points, 2 waves per WG, all GEMMs via WMMA.
// ---------------------------------------------------------------------------
__global__ __launch_bounds__(64) void fuse_kernel(
    const float* __restrict__ xt, const float* __restrict__ msg,
    const float* __restrict__ w_merge,
    const float* __restrict__ ln1_g, const float* __restrict__ ln1_b,
    const float* __restrict__ w_mlp1, const float* __restrict__ w_mlp2,
    const float* __restrict__ ln2_g, const float* __restrict__ ln2_b,
    float* __restrict__ out) {
  __shared__ float s_wm[32 * 33], s_w1[64 * 65], s_w2[64 * 33];
  __shared__ float s_x[2][16 * 33], s_m[2][16 * 33];
  __shared__ float s_cat[2][16 * 65], s_h[2][16 * 65];

  const int tid = threadIdx.x, lane = tid & 31, w = tid >> 5;
  for (int t = tid; t < 1024; t += 64) { const int r = t >> 5, c = t & 31; s_wm[r * 33 + c] = w_merge[t]; }
  for (int t = tid; t < 4096; t += 64) { const int r = t >> 6, c = t & 63; s_w1[r * 65 + c] = w_mlp1[t]; }
  for (int t = tid; t < 2048; t += 64) { const int r = t >> 5, c = t & 31; s_w2[r * 33 + c] = w_mlp2[t]; }

  const int p0 = (blockIdx.x * 2 + w) * 16;
  for (int t = lane; t < 16 * 32; t += 32) {
    const int r = t >> 5, c = t & 31;
    s_x[w][r * 33 + c] = xt[(size_t)(p0 + r) * CC + c];
    s_m[w][r * 33 + c] = msg[(size_t)(p0 + r) * CC + c];
  }
  __syncthreads();

  { // mm = msg @ w_merge  (compute both tiles, then overwrite s_m in place)
    v8f a0 = {};
    a0 = wmma_f32_16x16(&s_m[w][0], 33, 1, &s_wm[0], 33, 1, 32, a0);
    v8f a1 = {};
    a1 = wmma_f32_16x16(&s_m[w][0], 33, 1, &s_wm[16], 33, 1, 32, a1);
    const int cn = lane & 15, m0 = (lane >> 4) << 3;
    #pragma unroll
    for (int v = 0; v < 8; ++v) {
      s_m[w][(m0 + v) * 33 + cn] = a0[v];
      s_m[w][(m0 + v) * 33 + 16 + cn] = a1[v];
    }
  }
  __syncthreads();

  if (lane < 16) { // LN1 + concat [xt | ln1(mm)]
    const int r = lane;
    float mu = 0.f;
    for (int c = 0; c < 32; ++c) mu += s_m[w][r * 33 + c];
    mu *= (1.f / 32.f);
    float var = 0.f;
    for (int c = 0; c < 32; ++c) { const float d = s_m[w][r * 33 + c] - mu; var += d * d; }
    var *= (1.f / 32.f);
    const float ri = rsqrtf(var + LN_EPS);
    for (int c = 0; c < 32; ++c) {
      s_cat[w][r * 65 + c] = s_x[w][r * 33 + c];
      s_cat[w][r * 65 + 32 + c] = ln1_g[c] * (s_m[w][r * 33 + c] - mu) * ri + ln1_b[c];
    }
  }
  __syncthreads();

  // h = relu(cat @ w_mlp1)   [16x64]
  for (int nt = 0; nt < 64; nt += 16) {
    v8f a = {};
    a = wmma_f32_16x16(&s_cat[w][0], 65, 1, &s_w1[nt], 65, 1, 64, a);
    const int cn = lane & 15, m0 = (lane >> 4) << 3;
    #pragma unroll
    for (int v = 0; v < 8; ++v) s_h[w][(m0 + v) * 65 + nt + cn] = fmaxf(a[v], 0.f);
  }
  __syncthreads();

  // o = h @ w_mlp2   [16x32] -> reuse s_m
  for (int nt = 0; nt < 32; nt += 16) {
    v8f a = {};
    a = wmma_f32_16x16(&s_h[w][0], 65, 1, &s_w2[nt], 33, 1, 64, a);
    const int cn = lane & 15, m0 = (lane >> 4) << 3;
    #pragma unroll
    for (int v = 0; v < 8; ++v) s_m[w][(m0 + v) * 33 + nt + cn] = a[v];
  }
  __syncthreads();

  if (lane < 16) { // LN2 + residual + transposed store
    const int r = lane;
    const int p = p0 + r, b = p / NN, n = p % NN;
    float mu = 0.f;
    for (int c = 0; c < 32; ++c) mu += s_m[w][r * 33 + c];
    mu *= (1.f / 32.f);
    float var = 0.f;
    for (int c = 0; c < 32; ++c) { const float d = s_m[w][r * 33 + c] - mu; var += d * d; }
    var *= (1.f / 32.f);
    const float ri = rsqrtf(var + LN_EPS);
    for (int c = 0; c < 32; ++c) {
      const float y = ln2_g[c] * (s_m[w][r * 33 + c] - mu) * ri + ln2_b[c];
      out[(size_t)b * CC * NN + (size_t)c * NN + n] = s_x[w][r * 33 + c] + y;
    }
  }
}

// ---------------------------------------------------------------------------
extern "C" void kernel_launch(void* const* d_in, const int* in_sizes, int n_in,
                              void* d_out, int out_size, void* d_ws, size_t ws_size,
                              hipStream_t stream) {
  (void)in_sizes; (void)n_in; (void)out_size; (void)ws_size;
  const float* feat    = (const float*)d_in[0];
  const float* xyz     = (const float*)d_in[1];
  const float* w_pos1  = (const float*)d_in[2];
  const float* b_pos1  = (const float*)d_in[3];
  const float* w_pos2  = (const float*)d_in[4];
  const float* b_pos2  = (const float*)d_in[5];
  const float* w_q     = (const float*)d_in[6];
  const float* w_k     = (const float*)d_in[7];
  const float* w_v     = (const float*)d_in[8];
  const float* w_merge = (const float*)d_in[9];
  const float* ln1_g   = (const float*)d_in[10];
  const float* ln1_b   = (const float*)d_in[11];
  const float* w_mlp1  = (const float*)d_in[12];
  const float* w_mlp2  = (const float*)d_in[13];
  const float* ln2_g   = (const float*)d_in[14];
  const float* ln2_b   = (const float*)d_in[15];
  float* out = (float*)d_out;

  // workspace carve (~16.9 MB)
  float* xt = (float*)d_ws;
  float* xx = xt + (size_t)BB * NN * CC;
  float* qf = xx + (size_t)BB * NN;
  float* msg = qf + (size_t)BB * NN * CC;
  int* knn = (int*)(msg + (size_t)BB * NN * CC);

  const int npts = BB * NN;
  prep_kernel<<<npts / 256, 256, 0, stream>>>(feat, xyz, w_pos1, b_pos1, w_pos2, b_pos2,
                                              w_q, xt, xx, qf);
  knn_kernel<<<BB * (NN / QROWS), 128, 0, stream>>>(xt, xx, knn);
  attn_kernel<<<npts / 2, 64, 0, stream>>>(xt, xyz, knn, qf, w_pos1, b_pos1, w_pos2,
                                           b_pos2, w_k, w_v, msg);
  fuse_kernel<<<npts / 32, 64, 0, stream>>>(xt, msg, w_merge, ln1_g, ln1_b, w_mlp1,
                                            w_mlp2, ln2_g, ln2_b, out);
}